// ALIF_24309514895931
// MI455X (gfx1250) — compile-verified
//
#include <hip/hip_runtime.h>

#ifndef __has_builtin
#define __has_builtin(x) 0
#endif

#define BLOCK   256
#define STAGES  32           // LDS ring: 32 * 256 * 4B = 32 KB per block
#define UNROLL  8            // steps per sub-block; 4 sub-blocks = one ring revolution

typedef __attribute__((address_space(3))) float lds_f32;

// 32-bit LDS byte address of a generic pointer into __shared__ memory.
__device__ __forceinline__ unsigned lds_addr_of(float* p) {
    return (unsigned)(unsigned long long)(lds_f32*)p;
}

// CDNA5 async copy, GVS addressing: mem = SGPR_row(u64) + VGPR_off(i32).
// Per-lane 4B global -> LDS, tracked by ASYNCcnt; no VGPR destination,
// one shared lane-offset VGPR, row pointer stays on the scalar unit.
__device__ __forceinline__ void async_load_row(const float* row, unsigned lane_byte_off,
                                               unsigned lds_byte_addr) {
    asm volatile("global_load_async_to_lds_b32 %0, %1, %2"
                 :: "v"(lds_byte_addr), "v"(lane_byte_off), "s"(row)
                 : "memory");
}

template <int N>
__device__ __forceinline__ void wait_asynccnt() {
    asm volatile("s_wait_asynccnt %0" :: "n"(N) : "memory");
}

__device__ __forceinline__ void wait_dscnt0() {
    asm volatile("s_wait_dscnt 0" ::: "memory");
}

// One ALIF recurrence step; v and a live in the enclosing scope.
#define ALIF_STEP(XT, OUTP)                                             \
    {                                                                   \
        float u  = fmaf(decay_v, v, (XT));     /* leaky integrate */    \
        float th = fmaf(beta, a, threshold);   /* adaptive thresh */    \
        float d  = u - th;                                              \
        bool  sp = d > 0.0f;                                            \
        float s_ = sp ? 1.0f : 0.0f;                                    \
        v = sp ? d : u;                        /* v - s*th */           \
        a = fmaf(decay_a, a, s_);              /* adaptation */         \
        __builtin_nontemporal_store(s_, (OUTP));                        \
    }

// Fast path: requires n % BLOCK == 0 and T >= 2*STAGES.
__global__ __launch_bounds__(BLOCK) void alif_fwd_fast(
    const float* __restrict__ x,          // [T, n]
    const float* __restrict__ v0,
    const float* __restrict__ a0,
    const float* __restrict__ p_decay_v,
    const float* __restrict__ p_decay_a,
    const float* __restrict__ p_threshold,
    const float* __restrict__ p_beta,
    float* __restrict__ out,              // [T, n]
    int T, int n)
{
    __shared__ float xbuf[STAGES * BLOCK];

    const int tid = threadIdx.x;
    const int gi  = blockIdx.x * BLOCK + tid;   // dispatch guarantees gi < n

    float v = v0[gi];
    float a = a0[gi];
    const float decay_v   = p_decay_v[0];
    const float decay_a   = p_decay_a[0];
    const float threshold = p_threshold[0];
    const float beta      = p_beta[0];

    const unsigned lds0 = lds_addr_of(&xbuf[tid]);
    const unsigned gi4  = (unsigned)gi * 4u;    // lane byte offset, loop-invariant VGPR

    // Uniform (scalar) row cursors: address = row(SGPR) + gi(VGPR).
    const float* xrow = x;      // prefetch cursor: next time to issue
    float*       orow = out;    // output cursor: current time

    // ---- Prologue: issue times 0..STAGES-2 unconditionally (T >= 2*STAGES).
    #pragma unroll
    for (int s = 0; s < STAGES - 1; ++s) {
        async_load_row(xrow, gi4, lds0 + (unsigned)(s * BLOCK * 4));
        xrow += n;
    }
    // Invariant from here on: ops issued == t + STAGES - 1 (times 0..t+S-2).

    int t = 0;
    // ---- Main loop: one full ring revolution (32 steps) per body.
    //      All refills in bounds: t + 2*STAGES - 1 <= T.
    for (; t + 2 * STAGES - 1 <= T; t += STAGES) {
        #pragma unroll
        for (int j = 0; j < STAGES / UNROLL; ++j) {
            wait_asynccnt<STAGES - UNROLL - 1>();   // stages of this sub-block landed

            float xv[UNROLL];
            #pragma unroll
            for (int k = 0; k < UNROLL; ++k)        // compile-time LDS offsets
                xv[k] = xbuf[(j * UNROLL + k) * BLOCK + tid];
            wait_dscnt0();                          // drained -> slots reusable

            #pragma unroll
            for (int k = 0; k < UNROLL; ++k)        // refill: constant slots, uniform rows
                async_load_row(xrow + (size_t)k * n, gi4,
                    lds0 + (unsigned)((((j * UNROLL + STAGES - 1 + k) & (STAGES - 1)) * BLOCK) * 4));
            xrow += (size_t)UNROLL * n;

            #pragma unroll
            for (int k = 0; k < UNROLL; ++k)
                ALIF_STEP(xv[k], orow + (size_t)k * n + gi);
            orow += (size_t)UNROLL * n;
        }
    }

    // ---- Tail (a): keep issuing while time t+STAGES-1 < T  <=>  t <= T-STAGES.
    for (; t + STAGES <= T; ++t) {
        wait_asynccnt<STAGES - 2>();
        float xt = xbuf[(t & (STAGES - 1)) * BLOCK + tid];
        wait_dscnt0();
        async_load_row(xrow, gi4, lds0 + (unsigned)(((t + STAGES - 1) & (STAGES - 1)) * BLOCK * 4));
        xrow += n;
        ALIF_STEP(xt, orow + gi);
        orow += n;
    }

    // ---- Tail (b): exactly STAGES-1 steps remain; all T ops issued.
    //      Descending constant waits keep the drain pipelined.
    #define DRAIN_GROUP(WAITN, CNT)                                         \
        {                                                                   \
            wait_asynccnt<WAITN>();                                         \
            _Pragma("unroll")                                               \
            for (int k = 0; k < (CNT); ++k) {                               \
                float xt = xbuf[((t + k) & (STAGES - 1)) * BLOCK + tid];    \
                ALIF_STEP(xt, orow + (size_t)k * n + gi);                   \
            }                                                               \
            t += (CNT);                                                     \
            orow += (size_t)(CNT) * n;                                      \
        }
    DRAIN_GROUP(24, 7)   // steps T-31..T-25 : need completed >= T-24
    DRAIN_GROUP(16, 8)   // steps T-24..T-17 : need completed >= T-16
    DRAIN_GROUP(8,  8)   // steps T-16..T-9  : need completed >= T-8
    DRAIN_GROUP(0,  8)   // steps T-8 ..T-1  : need all complete
    #undef DRAIN_GROUP
}

// Generic fallback for shapes the fast path doesn't cover.
__global__ void alif_fwd_generic(
    const float* __restrict__ x,
    const float* __restrict__ v0,
    const float* __restrict__ a0,
    const float* __restrict__ p_decay_v,
    const float* __restrict__ p_decay_a,
    const float* __restrict__ p_threshold,
    const float* __restrict__ p_beta,
    float* __restrict__ out,
    int T, int n)
{
    const int gi = blockIdx.x * blockDim.x + threadIdx.x;
    if (gi >= n) return;

    float v = v0[gi];
    float a = a0[gi];
    const float decay_v   = p_decay_v[0];
    const float decay_a   = p_decay_a[0];
    const float threshold = p_threshold[0];
    const float beta      = p_beta[0];

    const float* xrow = x;
    float*       orow = out;
    for (int t = 0; t < T; ++t) {
        float xt = xrow[gi];
        ALIF_STEP(xt, orow + gi);
        xrow += n;
        orow += n;
    }
}

extern "C" void kernel_launch(void* const* d_in, const int* in_sizes, int n_in,
                              void* d_out, int out_size, void* d_ws, size_t ws_size,
                              hipStream_t stream) {
    // setup_inputs order: x_seq, v, a, decay_v, decay_a, threshold, beta, alpha
    const float* x   = (const float*)d_in[0];
    const float* v0  = (const float*)d_in[1];
    const float* a0  = (const float*)d_in[2];
    const float* dv  = (const float*)d_in[3];
    const float* da  = (const float*)d_in[4];
    const float* th  = (const float*)d_in[5];
    const float* be  = (const float*)d_in[6];
    // d_in[7] = alpha: only used by the backward surrogate, not the forward.
    float* out = (float*)d_out;

    const int n = in_sizes[1];            // B*N = 65536
    const int T = in_sizes[0] / n;        // 1024

    if ((n % BLOCK) == 0 && T >= 2 * STAGES) {
        alif_fwd_fast<<<n / BLOCK, BLOCK, 0, stream>>>(x, v0, a0, dv, da, th, be, out, T, n);
    } else {
        const int blocks = (n + BLOCK - 1) / BLOCK;
        alif_fwd_generic<<<blocks, BLOCK, 0, stream>>>(x, v0, a0, dv, da, th, be, out, T, n);
    }
}